// DecomLayer_74912819577632
// MI455X (gfx1250) — compile-verified
//
#include <hip/hip_runtime.h>
#include <math.h>

// Problem constants (fixed by the reference)
#define BB   64
#define NN   2048
#define MM   2048
#define CC   64
#define HH   128
#define NNZE 32768

typedef float v2f __attribute__((ext_vector_type(2)));
typedef float v8f __attribute__((ext_vector_type(8)));

// ---------------------------------------------------------------------------
// Kernel 1: per-graph SpMM (edge scatter) + cluster pooling, fused in LDS.
// grid.x = B * (H/32); block = 512 threads (16 waves, wave32).
// Dynamic LDS: coefs[M][32] | psum[C][32] | pmax[C][32] | cnt[C]  (~272 KB)
// 272 KB -> 1 WG per WGP, so 512 threads = 4 waves/SIMD32 for latency hiding.
// ---------------------------------------------------------------------------
__global__ __launch_bounds__(512)
void spmm_pool_kernel(const float* __restrict__ x,
                      const int*   __restrict__ rows,
                      const int*   __restrict__ cols,
                      const float* __restrict__ vals,
                      const int*   __restrict__ idx,
                      float* __restrict__ xdec) {
  extern __shared__ float smem[];
  float* sco  = smem;               // M*32
  float* psum = sco  + MM * 32;     // C*32
  float* pmax = psum + CC * 32;     // C*32
  float* cnt  = pmax + CC * 32;     // C

  const int tid    = threadIdx.x;
  const int lane   = tid & 31;
  const int wid    = tid >> 5;
  const int NWAVES = blockDim.x >> 5;
  const int b      = blockIdx.x >> 2;          // H/32 == 4 chunks
  const int hbase  = (blockIdx.x & 3) * 32;

  for (int i = tid; i < MM * 32; i += blockDim.x) sco[i] = 0.0f;
  for (int i = tid; i < CC * 32; i += blockDim.x) { psum[i] = 0.0f; pmax[i] = -INFINITY; }
  for (int i = tid; i < CC;      i += blockDim.x) cnt[i] = 0.0f;
  __syncthreads();

  // SpMM: each wave owns batches of 32 edges. Lane l loads edge (e0+l)'s
  // metadata once (fully coalesced), then the wave processes the 32 edges
  // one at a time with v_readlane broadcasts; within an edge, lane = feature
  // (coalesced 128B x gather, conflict-free ds_add_f32: 32 lanes -> 32 banks).
  const size_t ebase = (size_t)b * NNZE;
  const float* xg = x + (size_t)b * NN * HH + hbase;
  for (int e0 = wid * 32; e0 < NNZE; e0 += NWAVES * 32) {
    const int   rbuf = rows[ebase + e0 + lane];
    const int   cbuf = cols[ebase + e0 + lane];
    const float vbuf = vals[ebase + e0 + lane];
#pragma unroll
    for (int j = 0; j < 32; ++j) {          // j is constant -> v_readlane_b32
      const int   r = __shfl(rbuf, j, 32);
      const int   c = __shfl(cbuf, j, 32);
      const float v = __shfl(vbuf, j, 32);
      const float xv = xg[(size_t)c * HH + lane];
      atomicAdd(&sco[r * 32 + lane], v * xv);
    }
  }
  __syncthreads();

  // Cluster pooling (sum / max / count)
  const size_t ibase = (size_t)b * MM;
  for (int m = wid; m < MM; m += NWAVES) {
    const int   c = idx[ibase + m];
    const float v = sco[m * 32 + lane];
    atomicAdd(&psum[c * 32 + lane], v);
    __hip_atomic_fetch_max(&pmax[c * 32 + lane], v,
                           __ATOMIC_RELAXED, __HIP_MEMORY_SCOPE_WORKGROUP);
    if (lane == 0) atomicAdd(&cnt[c], 1.0f);
  }
  __syncthreads();

  // x_dec = [mean | max | sum]  ->  workspace, layout [B][C][3H]
  for (int c = wid; c < CC; c += NWAVES) {
    const float s    = psum[c * 32 + lane];
    const float mx   = pmax[c * 32 + lane];
    const float mean = s / fmaxf(cnt[c], 1.0f);
    const size_t base = ((size_t)b * CC + c) * (3 * HH);
    xdec[base +            hbase + lane] = mean;
    xdec[base +     HH +   hbase + lane] = mx;
    xdec[base + 2 * HH +   hbase + lane] = s;
  }
}

// ---------------------------------------------------------------------------
// Kernel 2: per-graph cluster self-attention on fp32 WMMA (16x16x4).
// grid.x = B; block = 256 threads (8 waves).
// LDS strides padded so 16-lane column reads hit distinct banks:
//   x_dec: 388 (388%64=4), Q/K/V: 132 (132%64=4), S: 66 (66%64=2)
// ---------------------------------------------------------------------------
#define XD_LD 388
#define QK_LD 132
#define S_LD  66

// A fragment (16x4 fp32): lanes 0-15 hold M rows with K={k,k+1},
// lanes 16-31 hold the same rows with K={k+2,k+3}.
__device__ __forceinline__ v2f ldsA(const float* base, int ld, int mt, int k, int lane) {
  const int hi = (lane >> 4) << 1;
  const float* p = base + (mt * 16 + (lane & 15)) * ld + k + hi;
  v2f a; a.x = p[0]; a.y = p[1];
  return a;
}

__global__ __launch_bounds__(256)
void cluster_attn_kernel(const float* __restrict__ xdec_g,
                         const float* __restrict__ Wq,
                         const float* __restrict__ Wk,
                         const float* __restrict__ Wv,
                         float* __restrict__ out) {
  extern __shared__ float smem[];
  float* xd = smem;                    // 64 * XD_LD
  float* Q  = xd + 64 * XD_LD;         // 64 * QK_LD
  float* K  = Q  + 64 * QK_LD;
  float* V  = K  + 64 * QK_LD;
  float* S  = xd;                      // 64 * S_LD, aliases dead x_dec region

  const int tid  = threadIdx.x;
  const int lane = tid & 31;
  const int wid  = tid >> 5;
  const int b    = blockIdx.x;

  // Stage x_dec [64][384] into padded LDS
  const float* src = xdec_g + (size_t)b * (CC * 3 * HH);
  for (int i = tid; i < CC * 3 * HH; i += blockDim.x)
    xd[(i / 384) * XD_LD + (i % 384)] = src[i];
  __syncthreads();

  // ---- Q/K/V = x_dec @ W : 96 16x16 tiles over 8 waves, K-dim 384 -------
  for (int t = wid; t < 96; t += 8) {
    const float* W = (t < 32) ? Wq : (t < 64) ? Wk : Wv;
    float*       D = (t < 32) ? Q  : (t < 64) ? K  : V;
    const int tl = t & 31, mt = tl >> 3, nt = tl & 7;
    const int hi = (lane >> 4) << 1, col = nt * 16 + (lane & 15);
    v8f acc = {};
    for (int k = 0; k < 3 * HH; k += 4) {
      v2f a = ldsA(xd, XD_LD, mt, k, lane);
      v2f bf; bf.x = W[(k + hi) * HH + col]; bf.y = W[(k + 1 + hi) * HH + col];
      acc = __builtin_amdgcn_wmma_f32_16x16x4_f32(false, a, false, bf,
                                                  (short)0, acc, false, false);
    }
    const int rbase = mt * 16 + ((lane >> 4) << 3);
#pragma unroll
    for (int r = 0; r < 8; ++r) D[(rbase + r) * QK_LD + col] = acc[r];
  }
  __syncthreads();

  // ---- S = (Q @ K^T) * 1/sqrt(H) : 16 tiles over 8 waves ----------------
  const float scale = 0.08838834764831845f;  // 1/sqrt(128)
  for (int t = wid; t < 16; t += 8) {
    const int mt = t >> 2, nt = t & 3;
    const int hi = (lane >> 4) << 1, col = nt * 16 + (lane & 15);
    v8f acc = {};
    for (int k = 0; k < HH; k += 4) {
      v2f a = ldsA(Q, QK_LD, mt, k, lane);
      v2f bf; bf.x = K[col * QK_LD + k + hi]; bf.y = K[col * QK_LD + k + hi + 1];
      acc = __builtin_amdgcn_wmma_f32_16x16x4_f32(false, a, false, bf,
                                                  (short)0, acc, false, false);
    }
    const int rbase = mt * 16 + ((lane >> 4) << 3);
#pragma unroll
    for (int r = 0; r < 8; ++r) S[(rbase + r) * S_LD + col] = acc[r] * scale;
  }
  __syncthreads();

  // ---- row softmax: wave per row, shuffle reductions --------------------
  for (int r = wid; r < CC; r += 8) {
    float v0 = S[r * S_LD + lane];
    float v1 = S[r * S_LD + 32 + lane];
    float m = fmaxf(v0, v1);
    for (int off = 16; off > 0; off >>= 1) m = fmaxf(m, __shfl_xor(m, off, 32));
    float e0 = expf(v0 - m), e1 = expf(v1 - m);
    float s = e0 + e1;
    for (int off = 16; off > 0; off >>= 1) s += __shfl_xor(s, off, 32);
    const float inv = 1.0f / s;
    S[r * S_LD + lane]      = e0 * inv;
    S[r * S_LD + 32 + lane] = e1 * inv;
  }
  __syncthreads();

  // ---- O = dist @ V : 32 tiles over 8 waves, write straight to d_out ----
  float* ob = out + (size_t)b * (CC * HH);
  for (int t = wid; t < 32; t += 8) {
    const int mt = t >> 3, nt = t & 7;
    const int hi = (lane >> 4) << 1, col = nt * 16 + (lane & 15);
    v8f acc = {};
    for (int k = 0; k < CC; k += 4) {
      v2f a = ldsA(S, S_LD, mt, k, lane);
      v2f bf; bf.x = V[(k + hi) * QK_LD + col]; bf.y = V[(k + 1 + hi) * QK_LD + col];
      acc = __builtin_amdgcn_wmma_f32_16x16x4_f32(false, a, false, bf,
                                                  (short)0, acc, false, false);
    }
    const int rbase = mt * 16 + ((lane >> 4) << 3);
#pragma unroll
    for (int r = 0; r < 8; ++r) ob[(rbase + r) * HH + col] = acc[r];
  }
}

// ---------------------------------------------------------------------------
extern "C" void kernel_launch(void* const* d_in, const int* in_sizes, int n_in,
                              void* d_out, int out_size, void* d_ws, size_t ws_size,
                              hipStream_t stream) {
  const float* x    = (const float*)d_in[0];
  // d_in[1] = batch (unused: nodes are sorted, equal graph sizes)
  // d_in[2] = batch_size scalar (compile-time constant BB)
  const int*   rows = (const int*)d_in[3];
  const int*   cols = (const int*)d_in[4];
  const float* vals = (const float*)d_in[5];
  const int*   idx  = (const int*)d_in[6];
  const float* Wq   = (const float*)d_in[7];
  const float* Wk   = (const float*)d_in[8];
  const float* Wv   = (const float*)d_in[9];
  float* xdec = (float*)d_ws;   // B*C*3H floats = 6 MB scratch

  const size_t smem1 = (size_t)(MM * 32 + 2 * CC * 32 + CC) * sizeof(float);
  spmm_pool_kernel<<<dim3(BB * (HH / 32)), dim3(512), smem1, stream>>>(
      x, rows, cols, vals, idx, xdec);

  const size_t smem2 = (size_t)(64 * XD_LD + 3 * 64 * QK_LD) * sizeof(float);
  cluster_attn_kernel<<<dim3(BB), dim3(256), smem2, stream>>>(
      xdec, Wq, Wk, Wv, (float*)d_out);
}